// ResidualAverageExtraBlock_41747082117851
// MI455X (gfx1250) — compile-verified
//
#include <hip/hip_runtime.h>

typedef __attribute__((ext_vector_type(2))) float v2f;
typedef __attribute__((ext_vector_type(8))) float v8f;

#define N_IN   128
#define N_MID  512
#define WS_WSUM   128   // ws[0..127] = weighted column sums, ws[128] = sum(w)
#define WS_OUTVEC 160   // ws[160..287] = o (broadcast residual vector)

// ---------------------------------------------------------------------------
// Pass 0: zero the accumulation workspace (must run every launch; the harness
// does not re-poison between graph replays and we accumulate via atomics).
// ---------------------------------------------------------------------------
__global__ void k_init(float* ws) {
    const int t = threadIdx.x;
    if (t < 320) ws[t] = 0.0f;
}

// ---------------------------------------------------------------------------
// Pass 1: ws[k] = sum_b w[b]*x[b,k]  (k<128),  ws[128] = sum_b w[b]
// One wave processes one full 512-byte row per iteration: lane L loads the
// float4 at columns 4L..4L+3 (global_load_b128, perfectly coalesced). Each
// lane owns fixed columns -> no cross-lane reduction; combine waves via LDS
// f32 atomics, then one set of global f32 atomics per block.
// ---------------------------------------------------------------------------
__global__ __launch_bounds__(256) void k_colsum(const float* __restrict__ x,
                                                const float* __restrict__ w,
                                                float* __restrict__ ws, int B) {
    __shared__ float s_col[N_IN];
    __shared__ float s_wsum;
    const int t    = threadIdx.x;
    const int lane = t & 31;
    const int wave = t >> 5;
    if (t < N_IN) s_col[t] = 0.0f;
    if (t == 0)   s_wsum   = 0.0f;
    __syncthreads();

    float4 acc = make_float4(0.f, 0.f, 0.f, 0.f);
    float  wsum = 0.f;
    const int stride = gridDim.x * 8;
    for (int r = blockIdx.x * 8 + wave; r < B; r += stride) {
        const float  wb = w[r];
        const size_t off = (size_t)r * N_IN + lane * 4;
        const float4 xv = *reinterpret_cast<const float4*>(x + off);
        __builtin_prefetch(x + off + (size_t)stride * N_IN, 0, 3);  // global_prefetch_b8
        acc.x = fmaf(wb, xv.x, acc.x);
        acc.y = fmaf(wb, xv.y, acc.y);
        acc.z = fmaf(wb, xv.z, acc.z);
        acc.w = fmaf(wb, xv.w, acc.w);
        if (lane == 0) wsum += wb;
    }
    atomicAdd(&s_col[lane * 4 + 0], acc.x);
    atomicAdd(&s_col[lane * 4 + 1], acc.y);
    atomicAdd(&s_col[lane * 4 + 2], acc.z);
    atomicAdd(&s_col[lane * 4 + 3], acc.w);
    if (lane == 0) atomicAdd(&s_wsum, wsum);
    __syncthreads();
    if (t < N_IN)  atomicAdd(&ws[t], s_col[t]);
    if (t == N_IN) atomicAdd(&ws[WS_WSUM], s_wsum);
}

// ---------------------------------------------------------------------------
// Pass 2 (single block, 8 waves): the two mat-vecs via V_WMMA_F32_16X16X4_F32.
//   h = relu(W_in @ [xbar; p])   (512x130)
//   o = W_out @ h                (128x512)
// A layout (16x4 f32): lane%16 = M row; VGPR0 holds K={0|2}, VGPR1 K={1|3}
// selected by lane half. B (4x16) mirrors that with the vector replicated
// across all 16 N columns, so every D column carries the mat-vec result.
// The p-term (K remainder) is folded into the C initializer.
// ---------------------------------------------------------------------------
__global__ __launch_bounds__(256) void k_mlp(float* ws, const float* __restrict__ p,
                                             const float* __restrict__ W_in,
                                             const float* __restrict__ W_out) {
    __shared__ float s_xbar[N_IN];
    __shared__ float s_h[N_MID];
    const int  t    = threadIdx.x;
    const int  lane = t & 31;
    const int  wave = t >> 5;
    const bool hi   = lane >= 16;
    const int  mrel = lane & 15;
    const int  k0   = hi ? 2 : 0;

    const float invw = 1.0f / ws[WS_WSUM];
    if (t < N_IN) s_xbar[t] = ws[t] * invw;
    __syncthreads();
    const float p0 = p[0], p1 = p[1];

    // ---- h = relu(W_in @ [xbar; p]) : 32 M-tiles, 4 per wave -------------
    for (int tile = wave; tile < N_MID / 16; tile += 8) {
        const int mbase = tile * 16;
        const int mhalf = mbase + (hi ? 8 : 0);
        v8f c;
#pragma unroll
        for (int r = 0; r < 8; ++r) {   // fold the 2 extra (p) columns into C
            const float* row = W_in + (size_t)(mhalf + r) * (N_IN + 2);
            c[r] = fmaf(row[N_IN], p0, row[N_IN + 1] * p1);
        }
        const float* Arow = W_in + (size_t)(mbase + mrel) * (N_IN + 2);
        for (int kb = 0; kb < N_IN; kb += 4) {
            v2f a; a.x = Arow[kb + k0];    a.y = Arow[kb + k0 + 1];
            v2f b; b.x = s_xbar[kb + k0];  b.y = s_xbar[kb + k0 + 1];
            c = __builtin_amdgcn_wmma_f32_16x16x4_f32(false, a, false, b,
                                                      (short)0, c, false, false);
        }
        if (mrel == 0) {                // lanes 0 and 16 hold column N=0
#pragma unroll
            for (int r = 0; r < 8; ++r) s_h[mhalf + r] = fmaxf(c[r], 0.0f);
        }
    }
    __syncthreads();

    // ---- o = W_out @ h : 8 M-tiles, 1 per wave ---------------------------
    {
        const int mbase = wave * 16;
        const int mhalf = mbase + (hi ? 8 : 0);
        v8f c = {0.f, 0.f, 0.f, 0.f, 0.f, 0.f, 0.f, 0.f};
        const float* Arow = W_out + (size_t)(mbase + mrel) * N_MID;
        for (int kb = 0; kb < N_MID; kb += 4) {
            v2f a; a.x = Arow[kb + k0];  a.y = Arow[kb + k0 + 1];
            v2f b; b.x = s_h[kb + k0];   b.y = s_h[kb + k0 + 1];
            c = __builtin_amdgcn_wmma_f32_16x16x4_f32(false, a, false, b,
                                                      (short)0, c, false, false);
        }
        if (mrel == 0) {
#pragma unroll
            for (int r = 0; r < 8; ++r) ws[WS_OUTVEC + mhalf + r] = c[r];
        }
    }
}

// ---------------------------------------------------------------------------
// Pass 3: out[b,:] = x[b,:] + o   (streaming, float4 per lane = 1 row/wave)
// ---------------------------------------------------------------------------
__global__ __launch_bounds__(256) void k_add(const float* __restrict__ x,
                                             const float* __restrict__ ws,
                                             float* __restrict__ out, int B) {
    const int t    = threadIdx.x;
    const int lane = t & 31;
    const int wave = t >> 5;
    const float4 ov = *reinterpret_cast<const float4*>(ws + WS_OUTVEC + lane * 4);
    const int stride = gridDim.x * 8;
    for (int r = blockIdx.x * 8 + wave; r < B; r += stride) {
        const size_t off = (size_t)r * N_IN + lane * 4;
        const float4 xv = *reinterpret_cast<const float4*>(x + off);
        float4 yv;
        yv.x = xv.x + ov.x; yv.y = xv.y + ov.y;
        yv.z = xv.z + ov.z; yv.w = xv.w + ov.w;
        *reinterpret_cast<float4*>(out + off) = yv;
    }
}

extern "C" void kernel_launch(void* const* d_in, const int* in_sizes, int n_in,
                              void* d_out, int out_size, void* d_ws, size_t ws_size,
                              hipStream_t stream) {
    const float* x     = (const float*)d_in[0];
    const float* w     = (const float*)d_in[1];
    const float* p     = (const float*)d_in[2];
    const float* W_in  = (const float*)d_in[3];
    const float* W_out = (const float*)d_in[4];
    float* out = (float*)d_out;
    float* ws  = (float*)d_ws;
    const int B = in_sizes[0] / N_IN;   // 262144

    k_init  <<<1,    320, 0, stream>>>(ws);
    k_colsum<<<512,  256, 0, stream>>>(x, w, ws, B);
    k_mlp   <<<1,    256, 0, stream>>>(ws, p, W_in, W_out);
    k_add   <<<2048, 256, 0, stream>>>(x, ws, out, B);
    (void)n_in; (void)out_size; (void)ws_size;
}